// YoloV1Loss_30855045054531
// MI455X (gfx1250) — compile-verified
//
#include <hip/hip_runtime.h>
#include <stdint.h>

#define NUM_CLASSES 20
#define CH 30                       // 20 classes + 2*(conf + 4 box)
#define TILE 256                    // cells per tile (== block size)
#define BLOCK 256                   // 8 waves (wave32)
#define ITERS 15                    // (TILE*CH/2) b64 chunks / BLOCK threads
#define TENSOR_FLOATS (TILE * CH)   // 7680 floats = 30720 B per tensor per buffer
#define TILE_BYTES (TILE * CH * 4)  // 30720 B
#define STEP_BYTES (BLOCK * 8)      // 2048 B advance per chunk iteration

// CDNA5 async global->LDS copy, GVS mode: SGPR base + 32-bit VGPR offset.
// Inline asm (no IOFFSET) => unambiguous semantics, portable across toolchains.
__device__ __forceinline__ void async_ld_b64_s(uint32_t lds_off, uint32_t voff,
                                               const void* sbase) {
    asm volatile("global_load_async_to_lds_b64 %0, %1, %2"
                 :: "v"(lds_off), "v"(voff), "s"(sbase)
                 : "memory");
}

__device__ __forceinline__ float iou_f(float tx, float ty, float tw, float th,
                                       float px, float py, float pw, float ph) {
    float t_x1 = tx - tw * 0.5f, t_y1 = ty - th * 0.5f;
    float t_x2 = tx + tw * 0.5f, t_y2 = ty + th * 0.5f;
    float p_x1 = px - pw * 0.5f, p_y1 = py - ph * 0.5f;
    float p_x2 = px + pw * 0.5f, p_y2 = py + ph * 0.5f;
    float x1 = fmaxf(t_x1, p_x1), y1 = fmaxf(t_y1, p_y1);
    float x2 = fminf(t_x2, p_x2), y2 = fminf(t_y2, p_y2);
    float inter = fmaxf(x2 - x1, 0.f) * fmaxf(y2 - y1, 0.f);
    float a1 = fabsf((t_x2 - t_x1) * (t_y2 - t_y1));
    float a2 = fabsf((p_x2 - p_x1) * (p_y2 - p_y1));
    return inter / (a1 + a2 - inter + 1e-6f);
}

__device__ __forceinline__ float cell_loss(const float* __restrict__ t,
                                           const float* __restrict__ p) {
    float obj = t[20];
    float tx = t[21], ty = t[22], tw = t[23], th = t[24];
    float c0 = p[20], x0 = p[21], y0 = p[22], w0 = p[23], h0 = p[24];
    float c1 = p[25], x1 = p[26], y1 = p[27], w1 = p[28], h1 = p[29];

    float i0 = iou_f(tx, ty, tw, th, x0, y0, w0, h0);
    float i1 = iou_f(tx, ty, tw, th, x1, y1, w1, h1);
    bool b1 = (i1 > i0);                 // argmax: ties -> box 0 (first index)

    float px = b1 ? x1 : x0;
    float py = b1 ? y1 : y0;
    float pw = b1 ? w1 : w0;
    float ph = b1 ? h1 : h0;
    float pc = b1 ? c1 : c0;
    float pi = b1 ? i1 : i0;

    float dx = tx - px, dy = ty - py;
    float xy = dx * dx + dy * dy;

    float sw = (pw > 0.f ? 1.f : 0.f) - (pw < 0.f ? 1.f : 0.f);   // jnp.sign
    float sh = (ph > 0.f ? 1.f : 0.f) - (ph < 0.f ? 1.f : 0.f);
    float ww = sqrtf(tw) - sw * sqrtf(fabsf(pw) + 1e-6f);
    float hh = sqrtf(th) - sh * sqrtf(fabsf(ph) + 1e-6f);
    float wh = ww * ww + hh * hh;

    float dconf = pi - pc;
    float objl = dconf * dconf;
    float noobjl = pc * pc;

    float cls = 0.f;
#pragma unroll
    for (int c = 0; c < NUM_CLASSES; ++c) {
        float d = t[c] - p[c];
        cls += d * d;
    }
    return obj * (5.0f * (xy + wh) + objl + cls) + 0.5f * (1.0f - obj) * noobjl;
}

__global__ void yolo_init_acc(double* acc) { *acc = 0.0; }

__global__ void yolo_finalize(const double* acc, float* out) { out[0] = (float)(*acc); }

__global__ __launch_bounds__(BLOCK)
void yolo_loss_main(const float* __restrict__ yt, const float* __restrict__ yp,
                    double* __restrict__ acc, long long ncells, long long fullTiles) {
    // [buffer][tensor][cell*30 + j] linear-in-chunk layout, 120 KB of 320 KB WGP LDS.
    // 64-bank LDS: stride-30-dword reads are conflict-free (30*d % 64 != 0 for d in 1..31).
    __shared__ float smem[2][2][TENSOR_FLOATS];
    __shared__ float red[BLOCK];

    const int tid = threadIdx.x;
    const uint32_t smemBase = (uint32_t)(uintptr_t)(&smem[0][0][0]);
    const uint32_t lane8 = (uint32_t)tid * 8u;

    const long long stride = gridDim.x;
    const long long tile0 = blockIdx.x;

    // Exactly 30 async instructions per wave per tile; addresses advance by a
    // constant 2048 B on both global and LDS sides (layout is linear in chunk id).
    auto issue_tile = [&](long long tile, int buf) {
        uint32_t voff = (uint32_t)(tile * (long long)TILE_BYTES) + lane8;
        uint32_t dT = smemBase + (uint32_t)((buf * 2 + 0) * (TENSOR_FLOATS * 4)) + lane8;
        uint32_t dP = smemBase + (uint32_t)((buf * 2 + 1) * (TENSOR_FLOATS * 4)) + lane8;
#pragma unroll
        for (int k = 0; k < ITERS; ++k) {
            async_ld_b64_s(dT, voff, (const void*)yt);
            async_ld_b64_s(dP, voff, (const void*)yp);
            voff += STEP_BYTES;
            dT += STEP_BYTES;
            dP += STEP_BYTES;
        }
    };

    float acc_local = 0.f;

    if (tile0 < fullTiles) issue_tile(tile0, 0);

    int it = 0;
    for (long long tile = tile0; tile < fullTiles; tile += stride, ++it) {
        const int buf = it & 1;
        const long long nxt = tile + stride;
        if (nxt < fullTiles) {
            issue_tile(nxt, buf ^ 1);                         // overlap next tile's DMA
            asm volatile("s_wait_asynccnt 30" ::: "memory");  // current tile's 30 done
        } else {
            asm volatile("s_wait_asynccnt 0" ::: "memory");
        }
        __syncthreads();

        const float* t = &smem[buf][0][tid * CH];
        const float* p = &smem[buf][1][tid * CH];
        acc_local += cell_loss(t, p);

        __syncthreads();   // protect this buffer before next iteration's prefetch rewrites it
    }

    // Tail cells (ncells % TILE, or everything if offsets exceed 32-bit): direct global path.
    if (blockIdx.x == 0) {
        for (long long cell = fullTiles * (long long)TILE + tid; cell < ncells; cell += BLOCK)
            acc_local += cell_loss(yt + cell * CH, yp + cell * CH);
    }

    // Block reduction, then one fp64 atomic per block (order-insensitive at fp32 precision).
    red[tid] = acc_local;
    __syncthreads();
#pragma unroll
    for (int s = BLOCK / 2; s > 0; s >>= 1) {
        if (tid < s) red[tid] += red[tid + s];
        __syncthreads();
    }
    if (tid == 0) atomicAdd(acc, (double)red[0]);
}

extern "C" void kernel_launch(void* const* d_in, const int* in_sizes, int n_in,
                              void* d_out, int out_size, void* d_ws, size_t ws_size,
                              hipStream_t stream) {
    (void)n_in; (void)out_size; (void)ws_size;
    const float* yt = (const float*)d_in[0];
    const float* yp = (const float*)d_in[1];
    double* acc = (double*)d_ws;

    long long total = (long long)in_sizes[0];        // floats per tensor
    long long ncells = total / CH;
    long long bytes = ncells * (long long)CH * 4ll;
    long long fullTiles = (bytes <= 0xFFFFFFFFll) ? (ncells / TILE) : 0ll;

    long long g = fullTiles < 1024 ? fullTiles : 1024;
    int grid = (int)(g < 1 ? 1 : g);

    yolo_init_acc<<<1, 1, 0, stream>>>(acc);
    yolo_loss_main<<<grid, BLOCK, 0, stream>>>(yt, yp, acc, ncells, fullTiles);
    yolo_finalize<<<1, 1, 0, stream>>>(acc, (float*)d_out);
}